// SinkhornOT_53257594470651
// MI455X (gfx1250) — compile-verified
//
#include <hip/hip_runtime.h>
#include <math.h>

typedef __attribute__((ext_vector_type(2))) float v2f;
typedef __attribute__((ext_vector_type(8))) float v8f;

#define DD 256
#define KK 128
#define MM 128
#define EPS_F 0.05f
#define ROWS_PER_BLOCK 32

static __device__ __forceinline__ v8f wmma_f32_k4(v2f a, v2f b, v8f c) {
  // D = A(16x4,f32) * B(4x16,f32) + C(16x16,f32)
  return __builtin_amdgcn_wmma_f32_16x16x4_f32(false, a, false, b, (short)0, c,
                                               false, false);
}

// ---------------------------------------------------------------------------
// Kernel 1: P = relu(X @ W1 + b1) @ W2 + b2   for 32 rows per block.
// 256 threads = 8 waves; each stage = 32 WMMA 16x16 tiles (2 row blk x 16 col).
// ---------------------------------------------------------------------------
__global__ __launch_bounds__(256) void mlp_kernel(
    const float* __restrict__ X, const float* __restrict__ W1,
    const float* __restrict__ b1, const float* __restrict__ W2,
    const float* __restrict__ b2, float* __restrict__ P) {
  __shared__ float Hs[ROWS_PER_BLOCK][DD];  // 32 KB

  const int tid   = threadIdx.x;
  const int lane  = tid & 31;
  const int wave  = tid >> 5;       // 0..7
  const int lhalf = lane >> 4;      // 0/1 -> K offset 0/2 in fragments
  const int lmod  = lane & 15;
  const int row0  = blockIdx.x * ROWS_PER_BLOCK;

  // ---- Stage 1: H = relu(X @ W1 + b1) -> LDS ----
  for (int t = 0; t < 4; ++t) {
    const int tile = wave * 4 + t;        // 0..31
    const int rb   = tile & 1;            // 16-row block
    const int n0   = (tile >> 1) * 16;    // 16-col block
    const int m0   = row0 + rb * 16;
    v8f acc = {};
    const float* Arow = X + (size_t)(m0 + lmod) * DD + 2 * lhalf;
    const float* Bcol = W1 + (size_t)(2 * lhalf) * DD + n0 + lmod;
    for (int k0 = 0; k0 < DD; k0 += 4) {
      v2f a; a.x = Arow[k0]; a.y = Arow[k0 + 1];
      v2f b; b.x = Bcol[(size_t)k0 * DD]; b.y = Bcol[(size_t)(k0 + 1) * DD];
      acc = wmma_f32_k4(a, b, acc);
    }
    const int n = n0 + lmod;
    const float bias = b1[n];
#pragma unroll
    for (int v = 0; v < 8; ++v) {
      const int mloc = rb * 16 + v + lhalf * 8;
      float val = acc[v] + bias;
      Hs[mloc][n] = val > 0.f ? val : 0.f;
    }
  }
  __syncthreads();

  // ---- Stage 2: P = H @ W2 + b2 -> global ----
  for (int t = 0; t < 4; ++t) {
    const int tile = wave * 4 + t;
    const int rb   = tile & 1;
    const int n0   = (tile >> 1) * 16;
    const int mloc0 = rb * 16;
    v8f acc = {};
    const float* Arow = &Hs[mloc0 + lmod][2 * lhalf];
    const float* Bcol = W2 + (size_t)(2 * lhalf) * DD + n0 + lmod;
    for (int k0 = 0; k0 < DD; k0 += 4) {
      v2f a; a.x = Arow[k0]; a.y = Arow[k0 + 1];
      v2f b; b.x = Bcol[(size_t)k0 * DD]; b.y = Bcol[(size_t)(k0 + 1) * DD];
      acc = wmma_f32_k4(a, b, acc);
    }
    const int n = n0 + lmod;
    const float bias = b2[n];
#pragma unroll
    for (int v = 0; v < 8; ++v) {
      const int mloc = mloc0 + v + lhalf * 8;
      P[(size_t)(row0 + mloc) * DD + n] = acc[v] + bias;
    }
  }
}

// ---------------------------------------------------------------------------
// Kernel 2: per batch b,  C = sqrt(max(qq + rr - 2*Q.R^T, 1e-6))  (128x128)
// 64 blocks x 256 threads; Gram via WMMA (B fragment = R row-major = R^T).
// ---------------------------------------------------------------------------
__global__ __launch_bounds__(256) void cost_kernel(const float* __restrict__ pq,
                                                   const float* __restrict__ pr,
                                                   float* __restrict__ Cout) {
  __shared__ float qq[KK];
  __shared__ float rr[MM];
  const int b = blockIdx.x;
  const float* Q = pq + (size_t)b * KK * DD;
  const float* R = pr + (size_t)b * MM * DD;
  const int tid = threadIdx.x;

  if (tid < KK) {
    float s = 0.f;
    const float* p = Q + (size_t)tid * DD;
    for (int d = 0; d < DD; ++d) s += p[d] * p[d];
    qq[tid] = s;
  } else {
    const int r = tid - KK;
    float s = 0.f;
    const float* p = R + (size_t)r * DD;
    for (int d = 0; d < DD; ++d) s += p[d] * p[d];
    rr[r] = s;
  }
  __syncthreads();

  const int lane = tid & 31, wave = tid >> 5;
  const int lhalf = lane >> 4, lmod = lane & 15;
  const size_t base = (size_t)b * KK * MM;
  for (int t = 0; t < 8; ++t) {
    const int tile = wave * 8 + t;        // 0..63
    const int m0 = (tile >> 3) * 16;
    const int n0 = (tile & 7) * 16;
    v8f acc = {};
    const float* Arow = Q + (size_t)(m0 + lmod) * DD + 2 * lhalf;
    const float* Brow = R + (size_t)(n0 + lmod) * DD + 2 * lhalf;  // B = R^T
    for (int k0 = 0; k0 < DD; k0 += 4) {
      v2f a;  a.x = Arow[k0];  a.y = Arow[k0 + 1];
      v2f bb; bb.x = Brow[k0]; bb.y = Brow[k0 + 1];
      acc = wmma_f32_k4(a, bb, acc);
    }
    const int n = n0 + lmod;
#pragma unroll
    for (int v = 0; v < 8; ++v) {
      const int m = m0 + v + lhalf * 8;
      const float d2 = qq[m] + rr[n] - 2.f * acc[v];
      Cout[base + (size_t)m * MM + n] = sqrtf(fmaxf(d2, 1e-6f));
    }
  }
}

// ---------------------------------------------------------------------------
// Kernel 3: Sinkhorn (10 iters) + mesh (3 iters) + outputs, one block/batch.
// 128 threads: thread t owns row t (row sweeps) and column t (col sweeps).
// Dynamic LDS: logK/T matrix 128x129 (padded) + small vectors (~68.6 KB of
// the 320 KB WGP pool).
// ---------------------------------------------------------------------------
__global__ __launch_bounds__(128) void sinkhorn_kernel(
    const float* __restrict__ mask_q, const float* __restrict__ mask_r,
    const float* __restrict__ Cmat, float* __restrict__ T_out,
    float* __restrict__ sig_out, float* __restrict__ cost_out) {
  extern __shared__ float smem[];
  const int LDP = MM + 1;
  float* lk  = smem;             // KK*LDP : log_K, later T
  float* lmu = lk + KK * LDP;    // KK
  float* lnu = lmu + KK;         // MM
  float* va  = lnu + MM;         // KK  (log_a)
  float* vb  = va + KK;          // MM  (log_b)
  float* red = vb + MM;          // 130 scratch

  const int b = blockIdx.x;
  const int t = threadIdx.x;     // 0..127

  // --- log-softmax of mask_q -> lmu ---
  const float xq = logf(fmaxf(mask_q[b * KK + t], 1e-8f));
  red[t] = xq;
  __syncthreads();
  if (t == 0) {
    float mx = red[0];
    for (int i = 1; i < KK; ++i) mx = fmaxf(mx, red[i]);
    float s = 0.f;
    for (int i = 0; i < KK; ++i) s += expf(red[i] - mx);
    red[128] = mx + logf(s);
  }
  __syncthreads();
  lmu[t] = xq - red[128];
  __syncthreads();

  // --- log-softmax of mask_r -> lnu ---
  const float xr = logf(fmaxf(mask_r[b * MM + t], 1e-8f));
  red[t] = xr;
  __syncthreads();
  if (t == 0) {
    float mx = red[0];
    for (int i = 1; i < MM; ++i) mx = fmaxf(mx, red[i]);
    float s = 0.f;
    for (int i = 0; i < MM; ++i) s += expf(red[i] - mx);
    red[128] = mx + logf(s);
  }
  __syncthreads();
  lnu[t] = xr - red[128];
  __syncthreads();

  // --- build log_K = -C/eps + lmu[row] + lnu[col] ---
  const size_t cbase = (size_t)b * KK * MM + (size_t)t * MM;
  for (int m = 0; m < MM; ++m)
    lk[t * LDP + m] = -Cmat[cbase + m] * (1.0f / EPS_F) + lmu[t] + lnu[m];
  va[t] = 0.f;
  vb[t] = 0.f;
  __syncthreads();

  // --- Sinkhorn iterations ---
  for (int it = 0; it < 10; ++it) {
    // log_a[t] = -LSE_m(lk[t][m] + vb[m])
    float mx = -3.402823466e38f;
    for (int m = 0; m < MM; ++m) mx = fmaxf(mx, lk[t * LDP + m] + vb[m]);
    float s = 0.f;
    for (int m = 0; m < MM; ++m) s += expf(lk[t * LDP + m] + vb[m] - mx);
    const float la = -(mx + logf(s));
    va[t] = la;
    __syncthreads();
    // log_b[t] = -LSE_k(lk[k][t] + va[k])
    mx = -3.402823466e38f;
    for (int k = 0; k < KK; ++k) mx = fmaxf(mx, lk[k * LDP + t] + va[k]);
    s = 0.f;
    for (int k = 0; k < KK; ++k) s += expf(lk[k * LDP + t] + va[k] - mx);
    vb[t] = -(mx + logf(s));
    __syncthreads();
  }

  // --- T = exp(log_K + log_a + log_b), stored in-place ---
  for (int m = 0; m < MM; ++m)
    lk[t * LDP + m] = expf(lk[t * LDP + m] + va[t] + vb[m]);
  __syncthreads();

  // --- mesh iterations: T = T*T; row-normalize; col-normalize ---
  for (int it = 0; it < 3; ++it) {
    float s = 0.f;
    for (int m = 0; m < MM; ++m) {
      float x = lk[t * LDP + m];
      x = x * x;
      lk[t * LDP + m] = x;
      s += x;
    }
    const float inv = 1.f / (s + 1e-8f);
    for (int m = 0; m < MM; ++m) lk[t * LDP + m] *= inv;
    __syncthreads();
    float cs = 0.f;
    for (int k = 0; k < KK; ++k) cs += lk[k * LDP + t];
    const float cinv = 1.f / (cs + 1e-8f);
    for (int k = 0; k < KK; ++k) lk[k * LDP + t] *= cinv;
    __syncthreads();
  }

  // --- cost = sum(T*C); outputs ---
  float ps = 0.f;
  for (int m = 0; m < MM; ++m) ps += lk[t * LDP + m] * Cmat[cbase + m];
  red[t] = ps;
  __syncthreads();
  if (t == 0) {
    float c = 0.f;
    for (int i = 0; i < KK; ++i) c += red[i];
    cost_out[b] = c;
    sig_out[b] = 1.f / (1.f + expf(c));  // sigmoid(-cost)
  }
  const size_t tbase = (size_t)b * KK * MM + (size_t)t * MM;
  for (int m = 0; m < MM; ++m) T_out[tbase + m] = lk[t * LDP + m];
}

// ---------------------------------------------------------------------------
extern "C" void kernel_launch(void* const* d_in, const int* in_sizes, int n_in,
                              void* d_out, int out_size, void* d_ws,
                              size_t ws_size, hipStream_t stream) {
  (void)in_sizes; (void)n_in; (void)out_size; (void)ws_size;
  const float* slots_q = (const float*)d_in[0];  // [64,128,256]
  const float* slots_r = (const float*)d_in[1];  // [64,128,256]
  const float* mask_q  = (const float*)d_in[2];  // [64,128]
  const float* mask_r  = (const float*)d_in[3];  // [64,128]
  const float* W1 = (const float*)d_in[4];       // [256,256]
  const float* b1 = (const float*)d_in[5];       // [256]
  const float* W2 = (const float*)d_in[6];       // [256,256]
  const float* b2 = (const float*)d_in[7];       // [256]

  // Output tuple layout: sigmoid [64] | T [64*128*128] | C [64*128*128] | cost [64]
  float* out      = (float*)d_out;
  float* sig_out  = out;
  float* T_out    = out + 64;
  float* C_out    = T_out + (size_t)64 * KK * MM;
  float* cost_out = C_out + (size_t)64 * KK * MM;

  // Workspace: pq (8 MB) | pr (8 MB)
  float* pq = (float*)d_ws;
  float* pr = pq + (size_t)64 * KK * DD;

  // MLP projections: 64*128 = 8192 rows each, 32 rows/block -> 256 blocks.
  mlp_kernel<<<256, 256, 0, stream>>>(slots_q, W1, b1, W2, b2, pq);
  mlp_kernel<<<256, 256, 0, stream>>>(slots_r, W1, b1, W2, b2, pr);

  // Cost matrix, one block per batch.
  cost_kernel<<<64, 256, 0, stream>>>(pq, pr, C_out);

  // Sinkhorn + mesh + outputs, one block per batch, dynamic LDS ~68.6 KB.
  const size_t smem_bytes =
      (size_t)(KK * (MM + 1) + KK + MM + KK + MM + 130) * sizeof(float);
  sinkhorn_kernel<<<64, 128, smem_bytes, stream>>>(mask_q, mask_r, C_out, T_out,
                                                   sig_out, cost_out);
}